// TTTBase_6356551598153
// MI455X (gfx1250) — compile-verified
//
#include <hip/hip_runtime.h>
#include <math.h>

// ---------------------------------------------------------------------------
// TTT layer for MI455X (gfx1250), fp32 via V_WMMA_F32_16X16X4_F32.
//  B=4, NM=256, MB=16, D=1024, NH=16, HD=64
//   1) proj_kernel : fused xq/xk/xv = x @ {wq,wk,wv}; LDS-tiled, async-DB GEMM
//   2) eta_kernel  : eta = clip(softplus(x@lr_kernel + b + base),0,1)
//   3) ttt_kernel  : sequential fast-weight scan, 1 WG/(b,h), 4 column-strip
//                    waves, async double-buffered tile staging
//   4) ln_kernel   : layernorm(Z) in place
//   5) out_kernel  : out = x + Z_ln @ wo ; LDS-tiled, async-DB GEMM
// ---------------------------------------------------------------------------

#define Bq   4
#define NM_  256
#define MB_  16
#define D_   1024
#define NH_  16
#define HD_  64
#define TOK_ (NM_ * MB_)      // 4096 tokens per batch
#define MTOT (Bq * TOK_)      // 16384 rows

#define BM 128                // GEMM block tile M
#define BN 64                 // GEMM block tile N (== HD)
#define BK 16                 // GEMM K chunk

typedef __attribute__((ext_vector_type(2))) float v2f;
typedef __attribute__((ext_vector_type(8))) float v8f;
typedef int i4vec __attribute__((vector_size(16)));
typedef __attribute__((address_space(1))) i4vec* as1_i4;
typedef __attribute__((address_space(3))) i4vec* as3_i4;

// D = A(16x4 f32) * B(4x16 f32) + C
static __device__ __forceinline__ v8f wmma_f32(v2f a, v2f b, v8f c) {
  return __builtin_amdgcn_wmma_f32_16x16x4_f32(false, a, false, b, (short)0, c,
                                               false, false);
}

// ---- async global -> LDS copy (ASYNCcnt path), with portable fallback ------
#if __has_builtin(__builtin_amdgcn_global_load_async_to_lds_b128)
static __device__ __forceinline__ void cp16(const float* g, float* l) {
  __builtin_amdgcn_global_load_async_to_lds_b128((as1_i4)g, (as3_i4)l, 0, 0);
}
static __device__ __forceinline__ void async_wait() {
#if __has_builtin(__builtin_amdgcn_s_wait_asynccnt)
  __builtin_amdgcn_s_wait_asynccnt(0);
#else
  asm volatile("s_wait_asynccnt 0x0" ::: "memory");
#endif
}
#else
static __device__ __forceinline__ void cp16(const float* g, float* l) {
  *(float4*)l = *(const float4*)g;
}
static __device__ __forceinline__ void async_wait() {}
#endif

// ---------------------------------------------------------------------------
// Shared GEMM block body: C[BM x BN] (x NW weight matrices) over K=1024.
// Block = 256 threads = 8 waves; wave (wm,wn) owns a 32x32 region (2x2 WMMA
// tiles). Double-buffered K-chunks staged in LDS via async copies.
// ---------------------------------------------------------------------------
template <int NW>
static __device__ __forceinline__ void gemm_block(
    const float* __restrict__ Abase, const float* const* Wm, int m0, int n0,
    int tid, v8f accs[NW][2][2]) {
  __shared__ alignas(16) float As[2][BM * BK];
  __shared__ alignas(16) float Bs[2][NW][BK * BN];

  const int lane = tid & 31, wid = tid >> 5;
  const int half = lane >> 4, l15 = lane & 15;
  const int wm = wid >> 1, wn = wid & 1;

  // copy mapping: A = 8 floats/thread (2xB128), B = 4 floats/thread/matrix
  const int arow = tid >> 1, ak = (tid & 1) * 8;
  const int brow = tid >> 4, bn = (tid & 15) * 4;

  // prologue: chunk 0 -> buffer 0
  {
    const float* ga = Abase + (size_t)(m0 + arow) * D_ + ak;
    cp16(ga, &As[0][arow * BK + ak]);
    cp16(ga + 4, &As[0][arow * BK + ak + 4]);
#pragma unroll
    for (int w = 0; w < NW; ++w)
      cp16(Wm[w] + (size_t)brow * D_ + n0 + bn, &Bs[0][w][brow * BN + bn]);
  }

  for (int kc = 0; kc < D_ / BK; ++kc) {
    const int cur = kc & 1;
    async_wait();          // my fills for chunk kc are done
    __syncthreads();       // everyone's fills done; buffers safe to swap
    if (kc + 1 < D_ / BK) {
      const int k0 = (kc + 1) * BK;
      const float* ga = Abase + (size_t)(m0 + arow) * D_ + k0 + ak;
      cp16(ga, &As[cur ^ 1][arow * BK + ak]);
      cp16(ga + 4, &As[cur ^ 1][arow * BK + ak + 4]);
#pragma unroll
      for (int w = 0; w < NW; ++w)
        cp16(Wm[w] + (size_t)(k0 + brow) * D_ + n0 + bn,
             &Bs[cur ^ 1][w][brow * BN + bn]);
    }
#pragma unroll
    for (int ks = 0; ks < 4; ++ks) {
      const int kk = ks * 4 + half * 2;
      v2f a[2];
#pragma unroll
      for (int mi = 0; mi < 2; ++mi) {
        const int r = wm * 32 + mi * 16 + l15;
        a[mi].x = As[cur][r * BK + kk];
        a[mi].y = As[cur][r * BK + kk + 1];
      }
#pragma unroll
      for (int w = 0; w < NW; ++w) {
        v2f bb[2];
#pragma unroll
        for (int ni = 0; ni < 2; ++ni) {
          const int c = wn * 32 + ni * 16 + l15;
          bb[ni].x = Bs[cur][w][kk * BN + c];
          bb[ni].y = Bs[cur][w][(kk + 1) * BN + c];
        }
#pragma unroll
        for (int mi = 0; mi < 2; ++mi)
#pragma unroll
          for (int ni = 0; ni < 2; ++ni)
            accs[w][mi][ni] = wmma_f32(a[mi], bb[ni], accs[w][mi][ni]);
      }
    }
  }
}

// ---------------------------------------------------------------------------
// 1) fused q/k/v projection. grid=(MTOT/BM=128, NH=16), block=256.
//    Output layout: XQ/XK/XV[((b*NH+h)*TOK + t) * HD + hd]
// ---------------------------------------------------------------------------
__global__ __launch_bounds__(256) void proj_kernel(
    const float* __restrict__ X, const float* __restrict__ Wq,
    const float* __restrict__ Wk, const float* __restrict__ Wv,
    float* __restrict__ XQ, float* __restrict__ XK, float* __restrict__ XV) {
  const int m0 = blockIdx.x * BM;
  const int h = blockIdx.y;          // BN == HD: one head per y-block
  const int n0 = h * BN;
  const int tid = threadIdx.x;
  const float* Ws[3] = {Wq, Wk, Wv};
  float* Outs[3] = {XQ, XK, XV};

  v8f accs[3][2][2] = {};
  gemm_block<3>(X, Ws, m0, n0, tid, accs);

  const int lane = tid & 31, wid = tid >> 5;
  const int half = lane >> 4, l15 = lane & 15;
  const int wm = wid >> 1, wn = wid & 1;
#pragma unroll
  for (int w = 0; w < 3; ++w) {
#pragma unroll
    for (int mi = 0; mi < 2; ++mi)
#pragma unroll
      for (int ni = 0; ni < 2; ++ni)
#pragma unroll
        for (int r = 0; r < 8; ++r) {
          const int mg = m0 + wm * 32 + mi * 16 + r + half * 8;
          const int b_ = mg / TOK_;
          const int t = mg % TOK_;
          const int dim = wn * 32 + ni * 16 + l15;
          Outs[w][(((size_t)b_ * NH_ + h) * (size_t)TOK_ + t) * HD_ + dim] =
              accs[w][mi][ni][r];
        }
  }
}

// ---------------------------------------------------------------------------
// 2) per-token learning rates. grid=1024, block=256. thread=(token,h).
// ---------------------------------------------------------------------------
__global__ __launch_bounds__(256) void eta_kernel(
    const float* __restrict__ X, const float* __restrict__ lrk,
    const float* __restrict__ lrb, const float* __restrict__ base,
    float* __restrict__ ETA) {
  const int t = blockIdx.x * 256 + threadIdx.x;
  const int h = t & 15;
  const int row = t >> 4;
  const float* __restrict__ x = X + (size_t)row * D_;
  float s = 0.f;
  for (int k = 0; k < D_; ++k) s = fmaf(x[k], lrk[k * NH_ + h], s);
  s += lrb[h] + base[0];
  float sp = (s > 20.f) ? s : log1pf(expf(s));
  sp = fminf(fmaxf(sp, 0.f), 1.0f);
  const int b_ = row / TOK_, tt = row % TOK_;
  const int nm = tt >> 4, tok = tt & 15;
  ETA[(((size_t)b_ * NH_ + h) * NM_ + nm) * MB_ + tok] = sp;
}

// ---------------------------------------------------------------------------
// 3) TTT fast-weight scan. grid = B*NH = 64, block = 128 (4 waves).
//    Wave w owns cols [w*16, w*16+16) of W[64,64] as 16 B-fragments.
//    Per-step tiles double-buffered in LDS, filled by async copies.
// ---------------------------------------------------------------------------
__global__ __launch_bounds__(128) void ttt_kernel(
    const float* __restrict__ XQ, const float* __restrict__ XK,
    const float* __restrict__ XV, const float* __restrict__ ETA,
    const unsigned char* __restrict__ mask, const float* __restrict__ W0,
    const float* __restrict__ b0, float* __restrict__ Zbuf,
    float* __restrict__ Wf_out, float* __restrict__ bf_out) {
  __shared__ alignas(16) float xq_s[2][MB_ * HD_];
  __shared__ alignas(16) float xk_s[2][MB_ * HD_];
  __shared__ alignas(16) float xv_s[2][MB_ * HD_];
  __shared__ float eta_s[MB_];
  __shared__ float e_s[4][MB_ * 16];
  __shared__ float gw_s[4][HD_ * 16];

  const int bh = blockIdx.x;
  const int b_ = bh >> 4, h = bh & 15;
  const int tid = threadIdx.x;
  const int wid = tid >> 5, lane = tid & 31;
  const int half = lane >> 4, l15 = lane & 15;
  const int ns = wid * 16;

  const float* __restrict__ xq = XQ + (size_t)bh * TOK_ * HD_;
  const float* __restrict__ xk = XK + (size_t)bh * TOK_ * HD_;
  const float* __restrict__ xv = XV + (size_t)bh * TOK_ * HD_;
  const float* __restrict__ ep = ETA + (size_t)bh * TOK_;

  auto issue = [&](int nm, int buf) {
    const size_t off = (size_t)nm * MB_ * HD_ + (size_t)tid * 8;
    cp16(xq + off, &xq_s[buf][tid * 8]);
    cp16(xq + off + 4, &xq_s[buf][tid * 8 + 4]);
    cp16(xk + off, &xk_s[buf][tid * 8]);
    cp16(xk + off + 4, &xk_s[buf][tid * 8 + 4]);
    cp16(xv + off, &xv_s[buf][tid * 8]);
    cp16(xv + off + 4, &xv_s[buf][tid * 8 + 4]);
  };

  // W strip -> B-fragments: frag kt holds rows kt*4 + half*2 + {0,1}.
  v2f wf[16];
  const size_t wbase = (size_t)h * HD_ * HD_;
#pragma unroll
  for (int kt = 0; kt < 16; ++kt) {
    const int k = kt * 4 + half * 2;
    wf[kt].x = W0[wbase + (size_t)k * HD_ + ns + l15];
    wf[kt].y = W0[wbase + (size_t)(k + 1) * HD_ + ns + l15];
  }
  float bvec = b0[h * HD_ + ns + l15];

  issue(0, 0);
  for (int nm = 0; nm < NM_; ++nm) {
    const int cur = nm & 1;
    async_wait();
    __syncthreads();
    if (nm + 1 < NM_) issue(nm + 1, cur ^ 1);   // overlap next step's fill
    if (tid < MB_) eta_s[tid] = ep[nm * MB_ + tid];
    const float m_val = (float)mask[b_ * NM_ + nm];

    // ---- err strip = xk_mb @ Wstrip  (16 tok x 16 col, K=64) ----
    v8f acc = {};
#pragma unroll
    for (int kt = 0; kt < 16; ++kt) {
      const int kk = kt * 4 + half * 2;
      v2f a;
      a.x = xk_s[cur][l15 * HD_ + kk];
      a.y = xk_s[cur][l15 * HD_ + kk + 1];
      acc = wmma_f32(a, wf[kt], acc);
    }
    __syncthreads();   // eta_s visible to all waves

    // e = eta * (err + b - xv); stash to LDS for re-fragmenting
#pragma unroll
    for (int r = 0; r < 8; ++r) {
      const int row = r + half * 8;
      const float e =
          eta_s[row] * (acc[r] + bvec - xv_s[cur][row * HD_ + ns + l15]);
      e_s[wid][row * 16 + l15] = e;
      acc[r] = e;
    }
    // gb = e.mean(0): 8 rows locally, other half-wave via shfl_xor(16)
    float gbp = 0.f;
#pragma unroll
    for (int r = 0; r < 8; ++r) gbp += acc[r];
    gbp += __shfl_xor(gbp, 16, 32);
    bvec -= m_val * (gbp * (1.0f / MB_));
    __syncthreads();

    // ---- gW strip = xk^T @ e  (64 dim x 16 col, K=16 tok) ----
#pragma unroll
    for (int mt2 = 0; mt2 < 4; ++mt2) {
      v8f g = {};
#pragma unroll
      for (int kt = 0; kt < 4; ++kt) {
        const int ktok = kt * 4 + half * 2;
        v2f a;   // A[m,k] = xk[k,m]
        a.x = xk_s[cur][ktok * HD_ + mt2 * 16 + l15];
        a.y = xk_s[cur][(ktok + 1) * HD_ + mt2 * 16 + l15];
        v2f bb;  // B[k,n] = e[k,n]
        bb.x = e_s[wid][ktok * 16 + l15];
        bb.y = e_s[wid][(ktok + 1) * 16 + l15];
        g = wmma_f32(a, bb, g);
      }
#pragma unroll
      for (int r = 0; r < 8; ++r)
        gw_s[wid][(mt2 * 16 + r + half * 8) * 16 + l15] = g[r];
    }
    __syncthreads();

    // ---- W strip update: W -= m * gW / MB ----
#pragma unroll
    for (int kt = 0; kt < 16; ++kt) {
      const int k = kt * 4 + half * 2;
      wf[kt].x -= m_val * gw_s[wid][k * 16 + l15] * (1.0f / MB_);
      wf[kt].y -= m_val * gw_s[wid][(k + 1) * 16 + l15] * (1.0f / MB_);
    }

    // ---- z strip = xq_mb @ W1 + b1 ----
    v8f zacc = {};
#pragma unroll
    for (int kt = 0; kt < 16; ++kt) {
      const int kk = kt * 4 + half * 2;
      v2f a;
      a.x = xq_s[cur][l15 * HD_ + kk];
      a.y = xq_s[cur][l15 * HD_ + kk + 1];
      zacc = wmma_f32(a, wf[kt], zacc);
    }
#pragma unroll
    for (int r = 0; r < 8; ++r) {
      const int row = r + half * 8;
      Zbuf[((size_t)b_ * TOK_ + (size_t)nm * MB_ + row) * D_ + h * HD_ + ns +
           l15] = zacc[r] + bvec;
    }
  }

  // final fast-weight state: Wf[b,h,:,:], bf[b,h,:]
  const size_t wout = (size_t)bh * HD_ * HD_;
#pragma unroll
  for (int kt = 0; kt < 16; ++kt) {
    const int k = kt * 4 + half * 2;
    Wf_out[wout + (size_t)k * HD_ + ns + l15] = wf[kt].x;
    Wf_out[wout + (size_t)(k + 1) * HD_ + ns + l15] = wf[kt].y;
  }
  if (half == 0) bf_out[(size_t)bh * HD_ + ns + l15] = bvec;
}

// ---------------------------------------------------------------------------
// 4) layernorm rows of Z in place. grid=16384, block=256.
// ---------------------------------------------------------------------------
__global__ __launch_bounds__(256) void ln_kernel(
    float* __restrict__ Z, const float* __restrict__ scale,
    const float* __restrict__ bias) {
  const int row = blockIdx.x;
  const int tid = threadIdx.x;
  float* __restrict__ z = Z + (size_t)row * D_;
  __shared__ float sred[8];

  float v[4];
  float s = 0.f;
#pragma unroll
  for (int i = 0; i < 4; ++i) {
    v[i] = z[tid + i * 256];
    s += v[i];
  }
#pragma unroll
  for (int o = 16; o > 0; o >>= 1) s += __shfl_xor(s, o, 32);
  if ((tid & 31) == 0) sred[tid >> 5] = s;
  __syncthreads();
  if (tid == 0) {
    float t = 0.f;
    for (int i = 0; i < 8; ++i) t += sred[i];
    sred[0] = t;
  }
  __syncthreads();
  const float mean = sred[0] * (1.0f / D_);
  __syncthreads();

  float vs = 0.f;
#pragma unroll
  for (int i = 0; i < 4; ++i) {
    const float d = v[i] - mean;
    vs += d * d;
  }
#pragma unroll
  for (int o = 16; o > 0; o >>= 1) vs += __shfl_xor(vs, o, 32);
  if ((tid & 31) == 0) sred[tid >> 5] = vs;
  __syncthreads();
  if (tid == 0) {
    float t = 0.f;
    for (int i = 0; i < 8; ++i) t += sred[i];
    sred[0] = t;
  }
  __syncthreads();
  const float rstd = rsqrtf(sred[0] * (1.0f / D_) + 1e-6f);
#pragma unroll
  for (int i = 0; i < 4; ++i) {
    const int c = tid + i * 256;
    z[c] = (v[i] - mean) * rstd * scale[c] + bias[c];
  }
}

// ---------------------------------------------------------------------------
// 5) out = x + Z_ln @ wo. grid=(128, 16), block=256.
// ---------------------------------------------------------------------------
__global__ __launch_bounds__(256) void out_kernel(
    const float* __restrict__ Zl, const float* __restrict__ Wo,
    const float* __restrict__ X, float* __restrict__ Out) {
  const int m0 = blockIdx.x * BM;
  const int n0 = blockIdx.y * BN;
  const int tid = threadIdx.x;
  const float* Ws[1] = {Wo};

  v8f accs[1][2][2] = {};
  gemm_block<1>(Zl, Ws, m0, n0, tid, accs);

  const int lane = tid & 31, wid = tid >> 5;
  const int half = lane >> 4, l15 = lane & 15;
  const int wm = wid >> 1, wn = wid & 1;
#pragma unroll
  for (int mi = 0; mi < 2; ++mi)
#pragma unroll
    for (int ni = 0; ni < 2; ++ni)
#pragma unroll
      for (int r = 0; r < 8; ++r) {
        const size_t mg = (size_t)m0 + wm * 32 + mi * 16 + r + half * 8;
        const size_t c = (size_t)n0 + wn * 32 + ni * 16 + l15;
        Out[mg * D_ + c] = X[mg * D_ + c] + accs[0][mi][ni][r];
      }
}

// ---------------------------------------------------------------------------
extern "C" void kernel_launch(void* const* d_in, const int* in_sizes, int n_in,
                              void* d_out, int out_size, void* d_ws,
                              size_t ws_size, hipStream_t stream) {
  const float* X = (const float*)d_in[0];
  const unsigned char* mask = (const unsigned char*)d_in[1];  // bool array
  const float* W0 = (const float*)d_in[2];
  const float* b0 = (const float*)d_in[3];
  const float* wq = (const float*)d_in[4];
  const float* wk = (const float*)d_in[5];
  const float* wv = (const float*)d_in[6];
  const float* wo = (const float*)d_in[7];
  const float* lrk = (const float*)d_in[8];
  const float* lrb = (const float*)d_in[9];
  const float* base = (const float*)d_in[10];
  const float* pscale = (const float*)d_in[11];
  const float* pbias = (const float*)d_in[12];

  float* out = (float*)d_out;                        // [4, 4096, 1024]
  float* Wf = out + (size_t)MTOT * D_;               // [4, 16, 64, 64]
  float* bf = Wf + (size_t)Bq * NH_ * HD_ * HD_;     // [4, 16, 64]

  float* ws = (float*)d_ws;
  float* XQ = ws;
  float* XK = XQ + (size_t)MTOT * D_;
  float* XV = XK + (size_t)MTOT * D_;
  float* ETA = XV + (size_t)MTOT * D_;               // 262144 floats
  float* Z = ETA + (size_t)Bq * NH_ * TOK_;

  proj_kernel<<<dim3(MTOT / BM, NH_), 256, 0, stream>>>(X, wq, wk, wv, XQ, XK,
                                                        XV);
  eta_kernel<<<(MTOT * NH_) / 256, 256, 0, stream>>>(X, lrk, lrb, base, ETA);
  ttt_kernel<<<Bq * NH_, 128, 0, stream>>>(XQ, XK, XV, ETA, mask, W0, b0, Z,
                                           Wf, bf);
  ln_kernel<<<MTOT, 256, 0, stream>>>(Z, pscale, pbias);
  out_kernel<<<dim3(MTOT / BM, D_ / BN), 256, 0, stream>>>(Z, wo, X, out);
}